// graph_net_44830868635842
// MI455X (gfx1250) — compile-verified
//
#include <hip/hip_runtime.h>

// ---------------------------------------------------------------------------
// The entire reference is affine in (cart, neigh): no activation anywhere, the
// neighbor mean is linear, the k comb rounds are linear, the node-sum and the
// decode are linear.  Hence
//   out = W_deco [ S*(Wm*m_sum + NODES*b_comb) + P*h_sum ] + b_deco
// with  P = Wh^k,  S = sum_{j<k} Wh^j,   Wh = W_comb[:,10:20], Wm = W_comb[:,0:10]
//   h_sum = (W_c2 W_c1) * Sum(cart)      + NODES*(W_c2 b_c1 + b_c2)
//   m_sum = (W_n2 W_n1) * Sum(neigh)/4   + NODES*(W_n2 b_n1 + b_n2)
// The only bulk work is streaming 44 MB -> 5 floats (memory-bound, ~1.9us at
// 23.3 TB/s).  The surviving matrix algebra (10x10 products / powers of Wh) is
// done with exact-f32 V_WMMA_F32_16X16X4_F32 in a single-wave kernel.
// ---------------------------------------------------------------------------

typedef __attribute__((ext_vector_type(2))) float v2f;
typedef __attribute__((ext_vector_type(4))) float f4;
typedef __attribute__((ext_vector_type(8))) float v8f;

#define NBLK 1024
#define TPB  256

// ---------------------------------------------------------------------------
// Kernel 1: bandwidth-bound streaming reduction.
//   cart  [NODES,3] -> per-component sums (component = flat_idx % 3)
//   neigh [NODES,4,2] -> even/odd component sums (component = flat_idx % 2)
// Deterministic: fixed grid, per-block partials written to d_ws.
// ---------------------------------------------------------------------------
__global__ __launch_bounds__(TPB) void reduce_sums(
    const float* __restrict__ cart, long ncart,
    const float* __restrict__ neigh, long nneigh,
    float* __restrict__ partials /* NBLK*5 */) {
  __shared__ float red[TPB];
  float acc[5] = {0.f, 0.f, 0.f, 0.f, 0.f};

  long tid    = (long)blockIdx.x * blockDim.x + threadIdx.x;
  long stride = (long)gridDim.x * blockDim.x;

  // cart: vector b128 nontemporal stream. float4 at 4*i has components
  // (4i)%3, (4i+1)%3, (4i+2)%3, (4i+3)%3 = c0, c0+1, c0+2, c0 with c0 = i%3.
  long n4c = ncart >> 2;
  const f4* c4 = reinterpret_cast<const f4*>(cart);
  for (long i = tid; i < n4c; i += stride) {
    f4 v = __builtin_nontemporal_load(&c4[i]);
    int c0 = (int)(i % 3);
    int c1 = c0 + 1; if (c1 == 3) c1 = 0;
    int c2 = c1 + 1; if (c2 == 3) c2 = 0;
    acc[c0] += v.x + v.w;
    acc[c1] += v.y;
    acc[c2] += v.z;
  }
  if (tid == 0)  // tail (none for 3,000,000, kept for generality)
    for (long i = n4c * 4; i < ncart; ++i) acc[(int)(i % 3)] += cart[i];

  // neigh: components alternate 0,1,0,1 inside each float4.
  long n4n = nneigh >> 2;
  const f4* g4 = reinterpret_cast<const f4*>(neigh);
  for (long i = tid; i < n4n; i += stride) {
    f4 v = __builtin_nontemporal_load(&g4[i]);
    acc[3] += v.x + v.z;
    acc[4] += v.y + v.w;
  }
  if (tid == 0)
    for (long i = n4n * 4; i < nneigh; ++i) acc[3 + (int)(i & 1)] += neigh[i];

  // block-level tree reduction, 5 components
  for (int c = 0; c < 5; ++c) {
    red[threadIdx.x] = acc[c];
    __syncthreads();
    for (int off = TPB / 2; off > 0; off >>= 1) {
      if ((int)threadIdx.x < off) red[threadIdx.x] += red[threadIdx.x + off];
      __syncthreads();
    }
    if (threadIdx.x == 0) partials[(long)blockIdx.x * 5 + c] = red[0];
    __syncthreads();
  }
}

// ---------------------------------------------------------------------------
// Kernel 2: one wave32.  Exact-f32 WMMA (V_WMMA_F32_16X16X4_F32) computes the
// collapsed weight matrices (W_c2@W_c1, W_n2@W_n1) and the k-dependent matrix
// polynomial P = Wh^k, S = sum_{j<k} Wh^j on 16x16 zero-padded tiles, then a
// tiny scalar epilogue folds in the reduced sums and writes the scalar output.
//
// f32 WMMA 16x16x4 layouts (ISA 7.12.2), lane = hi*16 + r (r=lane&15):
//   A (16x4):  A[r][kbase + 2*hi + {0,1}] in the v2f {x,y}
//   B (4x16):  B[kbase + 2*hi + {0,1}][r] in the v2f {x,y}
//   C/D 16x16: element [v + 8*hi][r] in C[v], v = 0..7
// ---------------------------------------------------------------------------
__global__ __launch_bounds__(32) void combine(
    const float* __restrict__ partials,
    const float* __restrict__ W_c1, const float* __restrict__ b_c1,
    const float* __restrict__ W_c2, const float* __restrict__ b_c2,
    const float* __restrict__ W_n1, const float* __restrict__ b_n1,
    const float* __restrict__ W_n2, const float* __restrict__ b_n2,
    const float* __restrict__ W_comb, const float* __restrict__ b_comb,
    const float* __restrict__ W_deco, const float* __restrict__ b_deco,
    const int* __restrict__ kptr, float nodes, float* __restrict__ out) {
  __shared__ float lsum[8];
  __shared__ float ldsP[256];   // 16x16 staging for C-layout -> B-layout
  __shared__ float ldsS[256];
  __shared__ float ldsAh[256];
  __shared__ float ldsAm[256];

  const int lane = threadIdx.x;
  const int r  = lane & 15;
  const int hi = lane >> 4;

  // fold per-block partials (fixed order -> deterministic)
  if (lane < 5) {
    float s = 0.f;
    for (int b = 0; b < NBLK; ++b) s += partials[b * 5 + lane];
    lsum[lane] = s;
  }
  __syncthreads();

  const int k = __builtin_amdgcn_readfirstlane(kptr[0]);  // scalar loop bound

  // A-layout chunks (K = 0..11, zero-padded past 10) for Wh, W_c2, W_n2
  v2f aWh[3], aC2[3], aN2[3];
  for (int c3 = 0; c3 < 3; ++c3) {
    int k0 = 4 * c3 + 2 * hi;
    aWh[c3].x = (r < 10 && k0     < 10) ? W_comb[r * 20 + 10 + k0    ] : 0.f;
    aWh[c3].y = (r < 10 && k0 + 1 < 10) ? W_comb[r * 20 + 10 + k0 + 1] : 0.f;
    aC2[c3].x = (r < 10 && k0     < 10) ? W_c2[r * 10 + k0    ] : 0.f;
    aC2[c3].y = (r < 10 && k0 + 1 < 10) ? W_c2[r * 10 + k0 + 1] : 0.f;
    aN2[c3].x = (r < 10 && k0     < 10) ? W_n2[r * 10 + k0    ] : 0.f;
    aN2[c3].y = (r < 10 && k0 + 1 < 10) ? W_n2[r * 10 + k0 + 1] : 0.f;
  }

  // Ah = W_c2 @ W_c1 (10x3), Am = W_n2 @ W_n1 (10x2): B chunks from global
  v8f Ah = {0.f, 0.f, 0.f, 0.f, 0.f, 0.f, 0.f, 0.f};
  v8f Am = {0.f, 0.f, 0.f, 0.f, 0.f, 0.f, 0.f, 0.f};
  for (int c3 = 0; c3 < 3; ++c3) {
    int row0 = 4 * c3 + 2 * hi;
    v2f b1, b2;
    b1.x = (row0     < 10 && r < 3) ? W_c1[(row0    ) * 3 + r] : 0.f;
    b1.y = (row0 + 1 < 10 && r < 3) ? W_c1[(row0 + 1) * 3 + r] : 0.f;
    Ah = __builtin_amdgcn_wmma_f32_16x16x4_f32(false, aC2[c3], false, b1,
                                               (short)0, Ah, false, false);
    b2.x = (row0     < 10 && r < 2) ? W_n1[(row0    ) * 2 + r] : 0.f;
    b2.y = (row0 + 1 < 10 && r < 2) ? W_n1[(row0 + 1) * 2 + r] : 0.f;
    Am = __builtin_amdgcn_wmma_f32_16x16x4_f32(false, aN2[c3], false, b2,
                                               (short)0, Am, false, false);
  }

  // P = Wh (C-layout), S = I; iterate P <- Wh@P, S <- S+P  (k-1 times)
  v8f P, S;
  for (int v = 0; v < 8; ++v) {
    int row = v + 8 * hi;
    P[v] = (row < 10 && r < 10) ? W_comb[row * 20 + 10 + r] : 0.f;
    S[v] = (row == r) ? 1.f : 0.f;
  }
  for (int j = 1; j < k; ++j) {
    for (int v = 0; v < 8; ++v) S[v] += P[v];
    for (int v = 0; v < 8; ++v) ldsP[(v + 8 * hi) * 16 + r] = P[v];  // C-layout out
    __syncthreads();
    v8f Pn = {0.f, 0.f, 0.f, 0.f, 0.f, 0.f, 0.f, 0.f};
    for (int c3 = 0; c3 < 3; ++c3) {
      int row0 = 4 * c3 + 2 * hi;
      v2f b;                                   // B-layout in
      b.x = ldsP[(row0    ) * 16 + r];
      b.y = ldsP[(row0 + 1) * 16 + r];
      Pn = __builtin_amdgcn_wmma_f32_16x16x4_f32(false, aWh[c3], false, b,
                                                 (short)0, Pn, false, false);
    }
    __syncthreads();
    P = Pn;
  }

  // publish matrices for the scalar epilogue
  for (int v = 0; v < 8; ++v) {
    int idx = (v + 8 * hi) * 16 + r;
    ldsP[idx]  = P[v];
    ldsS[idx]  = S[v];
    ldsAh[idx] = Ah[v];
    ldsAm[idx] = Am[v];
  }
  __syncthreads();

  if (lane == 0) {
    float Scart[3] = {lsum[0], lsum[1], lsum[2]};
    float Snb[2]   = {lsum[3] * 0.25f, lsum[4] * 0.25f};  // mean over 4 neighbors
    float hsum[10], msum[10], qsum[10];
    for (int i = 0; i < 10; ++i) {
      float ch = b_c2[i], cm = b_n2[i];
      for (int j = 0; j < 10; ++j) {
        ch += W_c2[i * 10 + j] * b_c1[j];
        cm += W_n2[i * 10 + j] * b_n1[j];
      }
      float hv = nodes * ch, mv = nodes * cm;
      for (int j = 0; j < 3; ++j) hv += ldsAh[i * 16 + j] * Scart[j];
      for (int j = 0; j < 2; ++j) mv += ldsAm[i * 16 + j] * Snb[j];
      hsum[i] = hv; msum[i] = mv;
    }
    for (int i = 0; i < 10; ++i) {
      float q = nodes * b_comb[i];
      for (int j = 0; j < 10; ++j) q += W_comb[i * 20 + j] * msum[j];
      qsum[i] = q;
    }
    float o = b_deco[0];
    for (int i = 0; i < 10; ++i) {
      float yi = 0.f;
      for (int j = 0; j < 10; ++j)
        yi += ldsS[i * 16 + j] * qsum[j] + ldsP[i * 16 + j] * hsum[j];
      o += W_deco[i] * yi;
    }
    out[0] = o;
  }
}

// ---------------------------------------------------------------------------
extern "C" void kernel_launch(void* const* d_in, const int* in_sizes, int n_in,
                              void* d_out, int out_size, void* d_ws, size_t ws_size,
                              hipStream_t stream) {
  const float* cart   = (const float*)d_in[0];
  const float* neigh  = (const float*)d_in[1];
  const float* W_c1   = (const float*)d_in[2];
  const float* b_c1   = (const float*)d_in[3];
  const float* W_c2   = (const float*)d_in[4];
  const float* b_c2   = (const float*)d_in[5];
  const float* W_n1   = (const float*)d_in[6];
  const float* b_n1   = (const float*)d_in[7];
  const float* W_n2   = (const float*)d_in[8];
  const float* b_n2   = (const float*)d_in[9];
  const float* W_comb = (const float*)d_in[10];
  const float* b_comb = (const float*)d_in[11];
  const float* W_deco = (const float*)d_in[12];
  const float* b_deco = (const float*)d_in[13];
  const int*   kptr   = (const int*)d_in[14];

  long ncart  = in_sizes[0];
  long nneigh = in_sizes[1];
  long nodes  = ncart / 3;

  float* partials = (float*)d_ws;  // NBLK*5 floats, fully overwritten each call

  reduce_sums<<<NBLK, TPB, 0, stream>>>(cart, ncart, neigh, nneigh, partials);
  combine<<<1, 32, 0, stream>>>(partials,
                                W_c1, b_c1, W_c2, b_c2,
                                W_n1, b_n1, W_n2, b_n2,
                                W_comb, b_comb, W_deco, b_deco,
                                kptr, (float)nodes, (float*)d_out);
}